// SRNN_11845519802807
// MI455X (gfx1250) — compile-verified
//
#include <hip/hip_runtime.h>
#include <hip/hip_bf16.h>
#include <math.h>

typedef __attribute__((ext_vector_type(2))) float v2f;
typedef __attribute__((ext_vector_type(8))) float v8f;

#define BATCH 32
#define DIM   128      // D == H == 128
#define TFULL 4096
#define SCAN_PF 8      // prefetch distance (time steps) in the scan

// D = A(16x4,f32) * B(4x16,f32) + C(16x16,f32)
static __device__ __forceinline__ v8f wmma4(v2f a, v2f b, v8f c) {
  return __builtin_amdgcn_wmma_f32_16x16x4_f32(
      /*neg_a=*/false, a, /*neg_b=*/false, b,
      /*c_mod=*/(short)0, c, /*reuse_a=*/false, /*reuse_b=*/false);
}

// ---------------------------------------------------------------------------
// Generic WMMA GEMM:  OUT[r, :] = X1[r, :] @ W1 (+ avg_sc(Xavg)[r, :] @ W2) + b1 (+ b2)
// rows = T_l * BATCH.  x1_btd != 0 means X1 is the original input in (B,T,D) layout.
// Xavg (if non-null) is always the original (B,T,D) input; scaled row t uses x[t..t+sc-1].
// nt_out != 0 -> non-temporal stores (write-once outputs; keep L2 for ws buffers).
// ---------------------------------------------------------------------------
__global__ __launch_bounds__(256)
void wmma_gemm_kernel(const float* __restrict__ X1, int x1_btd,
                      const float* __restrict__ Xavg, int sc,
                      const float* __restrict__ W1, const float* __restrict__ W2,
                      const float* __restrict__ b1, const float* __restrict__ b2,
                      float* __restrict__ OUT, int rows, int nt_out)
{
  __shared__ float ldsA1[64 * DIM];
  __shared__ float ldsA2[64 * DIM];

  const int tid    = threadIdx.x;
  const int wave   = tid >> 5;
  const int lane   = tid & 31;
  const int laneHi = (lane >> 4) & 1;
  const int lane15 = lane & 15;
  const int row0   = blockIdx.x * 64;

  const float invsc = (sc > 0) ? (1.0f / (float)sc) : 0.0f;

  // ---- stage A tiles into LDS (float4 granularity) ----
#pragma unroll
  for (int j = 0; j < 8; ++j) {
    int q  = tid + j * 256;     // 0..2047
    int rL = q >> 5;            // local row 0..63
    int c4 = q & 31;            // float4 column
    int grow = row0 + rL;
    float4 a1 = make_float4(0.f, 0.f, 0.f, 0.f);
    float4 a2 = make_float4(0.f, 0.f, 0.f, 0.f);
    if (grow < rows) {
      int t = grow >> 5;        // row = t*BATCH + b
      int b = grow & 31;
      size_t off;
      if (x1_btd) off = (size_t)b * (TFULL * DIM) + (size_t)t * DIM + c4 * 4;
      else        off = (size_t)grow * DIM + c4 * 4;
      a1 = *(const float4*)(X1 + off);
      if (Xavg) {
        size_t ao = (size_t)b * (TFULL * DIM) + (size_t)t * DIM + c4 * 4;
        float sx = 0.f, sy = 0.f, sz = 0.f, sw = 0.f;
        for (int i = 0; i < sc; ++i) {
          float4 v = *(const float4*)(Xavg + ao + (size_t)i * DIM);
          sx += v.x; sy += v.y; sz += v.z; sw += v.w;
        }
        a2 = make_float4(sx * invsc, sy * invsc, sz * invsc, sw * invsc);
      }
    }
    *(float4*)(ldsA1 + rL * DIM + c4 * 4) = a1;
    *(float4*)(ldsA2 + rL * DIM + c4 * 4) = a2;
  }
  __syncthreads();

  const int col = wave * 16 + lane15;     // this wave's N column

  v8f cacc[4];
#pragma unroll
  for (int mt = 0; mt < 4; ++mt) cacc[mt] = (v8f){0.f,0.f,0.f,0.f,0.f,0.f,0.f,0.f};

  // ---- B fragments for W1 (held in registers) ----
  v2f bf[32];
#pragma unroll
  for (int kk = 0; kk < 32; ++kk) {
    int r0 = 4 * kk + laneHi * 2;
    v2f b; b.x = W1[r0 * DIM + col]; b.y = W1[(r0 + 1) * DIM + col];
    bf[kk] = b;
  }
#pragma unroll
  for (int mt = 0; mt < 4; ++mt) {
#pragma unroll
    for (int kk = 0; kk < 32; ++kk) {
      int mrow = mt * 16 + lane15;
      int k0   = 4 * kk + laneHi * 2;
      v2f a; a.x = ldsA1[mrow * DIM + k0]; a.y = ldsA1[mrow * DIM + k0 + 1];
      cacc[mt] = wmma4(a, bf[kk], cacc[mt]);
    }
  }

  if (W2) {
#pragma unroll
    for (int kk = 0; kk < 32; ++kk) {
      int r0 = 4 * kk + laneHi * 2;
      v2f b; b.x = W2[r0 * DIM + col]; b.y = W2[(r0 + 1) * DIM + col];
      bf[kk] = b;
    }
#pragma unroll
    for (int mt = 0; mt < 4; ++mt) {
#pragma unroll
      for (int kk = 0; kk < 32; ++kk) {
        int mrow = mt * 16 + lane15;
        int k0   = 4 * kk + laneHi * 2;
        v2f a; a.x = ldsA2[mrow * DIM + k0]; a.y = ldsA2[mrow * DIM + k0 + 1];
        cacc[mt] = wmma4(a, bf[kk], cacc[mt]);
      }
    }
  }

  float bias = b1 ? b1[col] : 0.f;
  if (b2) bias += b2[col];

#pragma unroll
  for (int mt = 0; mt < 4; ++mt) {
#pragma unroll
    for (int v = 0; v < 8; ++v) {
      int m = mt * 16 + v + laneHi * 8;
      int grow = row0 + m;
      if (grow < rows) {
        float val = cacc[mt][v] + bias;
        float* p = OUT + (size_t)grow * DIM + col;
        if (nt_out) __builtin_nontemporal_store(val, p);
        else        *p = val;
      }
    }
  }
}

// ---------------------------------------------------------------------------
// Sequential scan: preh holds pre (T_l,B,H); overwritten in place with h.
// grid = 2 blocks, each handles 16 batch rows. Wh B-fragments live in VGPRs,
// h lives in LDS; per step: 32 ds_load_b64 A-frags + 32 v_wmma + tanh.
// pre rows are prefetched SCAN_PF steps ahead (global_prefetch_b8) so the
// serial dependence chain sees L2/L0 hits instead of full-latency loads.
// ---------------------------------------------------------------------------
__global__ __launch_bounds__(256)
void scan_kernel(float* __restrict__ preh, const float* __restrict__ Wh,
                 const float* __restrict__ bh, const float* __restrict__ h0,
                 int Tl)
{
  __shared__ float hl[16 * DIM];   // this block's 16 batch rows of h

  const int tid    = threadIdx.x;
  const int wave   = tid >> 5;
  const int lane   = tid & 31;
  const int laneHi = (lane >> 4) & 1;
  const int lane15 = lane & 15;
  const int brow0  = blockIdx.x * 16;
  const int col    = wave * 16 + lane15;

  // resident Wh B-fragments (64 VGPRs)
  v2f bf[32];
#pragma unroll
  for (int kk = 0; kk < 32; ++kk) {
    int r0 = 4 * kk + laneHi * 2;
    v2f b; b.x = Wh[r0 * DIM + col]; b.y = Wh[(r0 + 1) * DIM + col];
    bf[kk] = b;
  }

  // init h from h0
#pragma unroll
  for (int j = 0; j < 8; ++j) {
    int q = tid * 8 + j;            // 0..2047
    int r = q >> 7, c = q & 127;
    hl[q] = h0[(size_t)(brow0 + r) * DIM + c];
  }
  const float bias = bh[col];
  __syncthreads();

  // this block's slice of a time step is contiguous: rows brow0..brow0+15
  const size_t sliceOff = (size_t)brow0 * DIM;   // + t*BATCH*DIM

  for (int t = 0; t < Tl; ++t) {
    float* prow = preh + (size_t)t * BATCH * DIM;

    // prefetch the pre slice SCAN_PF steps ahead (2048 floats / 256 threads)
    if (t + SCAN_PF < Tl) {
      const float* pf = preh + (size_t)(t + SCAN_PF) * BATCH * DIM + sliceOff + tid * 8;
      __builtin_prefetch(pf, 0, 3);
    }

    // pre_t elements matching this lane's C fragment
    float p[8];
#pragma unroll
    for (int v = 0; v < 8; ++v) {
      int m = v + laneHi * 8;
      p[v] = prow[(size_t)(brow0 + m) * DIM + col];
    }

    v8f c = (v8f){0.f,0.f,0.f,0.f,0.f,0.f,0.f,0.f};
#pragma unroll
    for (int kk = 0; kk < 32; ++kk) {
      int k0 = 4 * kk + laneHi * 2;
      v2f a; a.x = hl[lane15 * DIM + k0]; a.y = hl[lane15 * DIM + k0 + 1];
      c = wmma4(a, bf[kk], c);
    }
    __syncthreads();   // all waves finished reading old h

    float hn[8];
#pragma unroll
    for (int v = 0; v < 8; ++v) hn[v] = tanhf(c[v] + p[v] + bias);
#pragma unroll
    for (int v = 0; v < 8; ++v) {
      int m = v + laneHi * 8;
      hl[m * DIM + col] = hn[v];                        // for next step's A frags
      prow[(size_t)(brow0 + m) * DIM + col] = hn[v];    // hseq output (in place)
    }
    __syncthreads();   // h fully updated
  }
}

// ---------------------------------------------------------------------------
// Moving-average outputs sx_l (T_l,B,D) from input x in (B,T,D) layout.
// Non-temporal stores: write-once output, keep L2 for ws/x working set.
// ---------------------------------------------------------------------------
__global__ void sx_kernel(const float* __restrict__ x, float* __restrict__ out,
                          int Tl, int sc, float invsc)
{
  int e = blockIdx.x * 256 + threadIdx.x;
  int total = Tl * BATCH * DIM;
  if (e >= total) return;
  int d = e & 127;
  int b = (e >> 7) & 31;
  int t = e >> 12;
  const float* src = x + (size_t)b * (TFULL * DIM) + (size_t)t * DIM + d;
  float s = 0.f;
  for (int i = 0; i < sc; ++i) s += src[(size_t)i * DIM];
  __builtin_nontemporal_store(s * invsc, out + e);
}

extern "C" void kernel_launch(void* const* d_in, const int* in_sizes, int n_in,
                              void* d_out, int out_size, void* d_ws, size_t ws_size,
                              hipStream_t stream)
{
  (void)in_sizes; (void)n_in; (void)out_size; (void)ws_size;

  const float* x  = (const float*)d_in[0];   // (B,T,D)
  const float* Wi = (const float*)d_in[1];   // (L,D,H)
  const float* bi = (const float*)d_in[2];   // (L,H)
  const float* Ws = (const float*)d_in[3];   // (L,D,H)
  const float* bs = (const float*)d_in[4];   // (L,H)
  const float* Wh = (const float*)d_in[5];   // (L,H,H)
  const float* bh = (const float*)d_in[6];   // (L,H)
  const float* Wl = (const float*)d_in[7];   // (H,D)
  const float* bl = (const float*)d_in[8];   // (D,)
  const float* h0 = (const float*)d_in[9];   // (L,B,H)
  float* out = (float*)d_out;

  const int T0 = 4096, T1 = 4095, T2 = 4093;
  const size_t PER = (size_t)BATCH * DIM;    // elements per time step = 4096
  const int MAT = DIM * DIM;

  float* ws0 = (float*)d_ws;                 // 64 MB buffer A
  float* ws1 = ws0 + (size_t)T0 * PER;       // 64 MB buffer B

  const size_t off_out0 = 0;
  const size_t off_out1 = off_out0 + (size_t)T0 * PER;
  const size_t off_out2 = off_out1 + (size_t)T1 * PER;
  const size_t off_sx0  = off_out2 + (size_t)T2 * PER;
  const size_t off_sx1  = off_sx0  + (size_t)T0 * PER;
  const size_t off_sx2  = off_sx1  + (size_t)T1 * PER;

  // sx outputs (independent, bandwidth-bound)
  sx_kernel<<<(T0 * (int)PER + 255) / 256, 256, 0, stream>>>(x, out + off_sx0, T0, 1, 1.0f);
  sx_kernel<<<(T1 * (int)PER + 255) / 256, 256, 0, stream>>>(x, out + off_sx1, T1, 2, 0.5f);
  sx_kernel<<<(T2 * (int)PER + 255) / 256, 256, 0, stream>>>(x, out + off_sx2, T2, 4, 0.25f);

  // ---- layer 0 ----
  {
    int rows = T0 * BATCH;
    wmma_gemm_kernel<<<(rows + 63) / 64, 256, 0, stream>>>(
        x, 1, x, 1, Wi + 0 * MAT, Ws + 0 * MAT, bi + 0 * DIM, bs + 0 * DIM, ws0, rows, 0);
    scan_kernel<<<2, 256, 0, stream>>>(ws0, Wh + 0 * MAT, bh + 0 * DIM, h0 + 0 * BATCH * DIM, T0);
  }
  // ---- layer 1 pre (reads hseq0) + out0 projection ----
  {
    int rows = T1 * BATCH;
    wmma_gemm_kernel<<<(rows + 63) / 64, 256, 0, stream>>>(
        ws0, 0, x, 2, Wi + 1 * MAT, Ws + 1 * MAT, bi + 1 * DIM, bs + 1 * DIM, ws1, rows, 0);
    int rows0 = T0 * BATCH;
    wmma_gemm_kernel<<<(rows0 + 63) / 64, 256, 0, stream>>>(
        ws0, 0, nullptr, 0, Wl, nullptr, bl, nullptr, out + off_out0, rows0, 1);
    scan_kernel<<<2, 256, 0, stream>>>(ws1, Wh + 1 * MAT, bh + 1 * DIM, h0 + 1 * BATCH * DIM, T1);
  }
  // ---- layer 2 pre (reads hseq1 with offset 1 step) + out1 projection ----
  {
    int rows = T2 * BATCH;
    wmma_gemm_kernel<<<(rows + 63) / 64, 256, 0, stream>>>(
        ws1 + PER, 0, x, 4, Wi + 2 * MAT, Ws + 2 * MAT, bi + 2 * DIM, bs + 2 * DIM, ws0, rows, 0);
    int rows1 = T1 * BATCH;
    wmma_gemm_kernel<<<(rows1 + 63) / 64, 256, 0, stream>>>(
        ws1, 0, nullptr, 0, Wl, nullptr, bl, nullptr, out + off_out1, rows1, 1);
    scan_kernel<<<2, 256, 0, stream>>>(ws0, Wh + 2 * MAT, bh + 2 * DIM, h0 + 2 * BATCH * DIM, T2);
    int rows2 = T2 * BATCH;
    wmma_gemm_kernel<<<(rows2 + 63) / 64, 256, 0, stream>>>(
        ws0, 0, nullptr, 0, Wl, nullptr, bl, nullptr, out + off_out2, rows2, 1);
  }
}